// AtomFeature_18330920419810
// MI455X (gfx1250) — compile-verified
//
#include <hip/hip_runtime.h>

#define NMAIN 6
#define BSZ   4
#define NATM  3072
#define KNN   32
#define DIM   128
#define SEPSF 1e-6f
#define LEPSF 1e6f

#define ROWS    32      // rows per kNN block (one wave, one row per lane)
#define CCH     128     // column chunk
#define DSTRIDE 132     // LDS dist-tile row stride: 16B-aligned rows for b128 scans

typedef float v2f __attribute__((ext_vector_type(2)));
typedef float v8f __attribute__((ext_vector_type(8)));

// ---------------------------------------------------------------------------
// Graph-norm statistics: per (b,d) mean and 1/sqrt(var+eps), exploiting the
// fact that emb[b,n,:] = emb_table[n%6,:]*mask -> only need per-type counts.
// ---------------------------------------------------------------------------
__global__ __launch_bounds__(128) void graphnorm_stats(
    const float* __restrict__ mask, const float* __restrict__ table,
    float* __restrict__ stats) {
  const int b = blockIdx.x;
  const int tid = threadIdx.x;
  __shared__ float cnt[NMAIN];
  if (tid < NMAIN) cnt[tid] = 0.f;
  __syncthreads();
  float loc[NMAIN] = {0.f, 0.f, 0.f, 0.f, 0.f, 0.f};
  for (int n = tid; n < NATM; n += 128) loc[n % NMAIN] += mask[b * NATM + n];
  for (int t = 0; t < NMAIN; ++t)
    if (loc[t] != 0.f) atomicAdd(&cnt[t], loc[t]);
  __syncthreads();
  float total = 0.f;
  for (int t = 0; t < NMAIN; ++t) total += cnt[t];
  const float denom = fmaxf(total, 1.f);
  const int d = tid;  // blockDim == DIM == 128
  float mean = 0.f;
  for (int t = 0; t < NMAIN; ++t) mean += cnt[t] * table[t * DIM + d];
  mean /= denom;
  float var = 0.f;
  for (int t = 0; t < NMAIN; ++t) {
    const float df = table[t * DIM + d] - mean;
    var += cnt[t] * df * df;
  }
  var /= denom;
  stats[b * 2 * DIM + d]       = mean;
  stats[b * 2 * DIM + DIM + d] = 1.f / sqrtf(var + 1e-5f);
}

// ---------------------------------------------------------------------------
// Graph-norm apply: out[b,n,d] = ((T[n%6,d]*m - mean)*rstd*gamma + beta)*m
// ---------------------------------------------------------------------------
__global__ __launch_bounds__(128) void graphnorm_apply(
    const float* __restrict__ mask, const float* __restrict__ table,
    const float* __restrict__ gamma, const float* __restrict__ beta,
    const float* __restrict__ stats, float* __restrict__ out) {
  const int b  = blockIdx.y;
  const int n0 = blockIdx.x * 8;
  const int d  = threadIdx.x;
  const float mean = stats[b * 2 * DIM + d];
  const float rstd = stats[b * 2 * DIM + DIM + d];
  const float g  = gamma[d];
  const float be = beta[d];
#pragma unroll
  for (int r = 0; r < 8; ++r) {
    const int n = n0 + r;
    const float m = mask[b * NATM + n];
    const float e = table[(n % NMAIN) * DIM + d] * m;
    out[((size_t)b * NATM + n) * DIM + d] = ((e - mean) * rstd * g + be) * m;
  }
}

// ---------------------------------------------------------------------------
// kNN: one wave per 32 rows. WMMA f32 16x16x4 computes coord dot-product
// tiles (K=3 padded to 4); epilogue forms masked distances into an LDS tile;
// each lane then maintains a 32-entry insertion-sorted top-k list in LDS.
// ---------------------------------------------------------------------------
__global__ __launch_bounds__(32) void knn_kernel(
    const float* __restrict__ coords, const float* __restrict__ mask,
    float* __restrict__ dists_out, int* __restrict__ edge_out) {
  const int b   = blockIdx.y;
  const int r0  = blockIdx.x * ROWS;
  const int tid = threadIdx.x;       // single wave32
  const int m16 = tid & 15;
  const bool hi = tid >= 16;

  __shared__ float rowX[ROWS], rowY[ROWS], rowZ[ROWS], rowS[ROWS], rowM[ROWS];
  __shared__ float colX[CCH], colY[CCH], colZ[CCH], colS[CCH], colM[CCH];
  __shared__ __align__(16) float dtile[ROWS * DSTRIDE];
  __shared__ float bestD[KNN * ROWS];   // [pos][lane] layout: conflict-free
  __shared__ int   bestI[KNN * ROWS];

  // stage the block's 32 row coordinates (one per lane)
  {
    const size_t cb = ((size_t)b * NATM + (r0 + tid)) * 3;
    const float x = coords[cb + 0], y = coords[cb + 1], z = coords[cb + 2];
    rowX[tid] = x; rowY[tid] = y; rowZ[tid] = z;
    rowS[tid] = x * x + y * y + z * z;
    rowM[tid] = mask[b * NATM + r0 + tid];
  }
#pragma unroll
  for (int p = 0; p < KNN; ++p) {
    bestD[p * ROWS + tid] = 3.0e38f;
    bestI[p * ROWS + tid] = -1;
  }
  float kworst = 3.0e38f;
  __syncthreads();

  for (int c0 = 0; c0 < NATM; c0 += CCH) {
    // stage column chunk
    for (int t = tid; t < CCH; t += 32) {
      const size_t cb = ((size_t)b * NATM + (c0 + t)) * 3;
      const float x = coords[cb + 0], y = coords[cb + 1], z = coords[cb + 2];
      colX[t] = x; colY[t] = y; colZ[t] = z;
      colS[t] = x * x + y * y + z * z;
      colM[t] = mask[b * NATM + c0 + t];
    }
    __syncthreads();

    // --- phase A: WMMA distance tiles -> LDS -------------------------------
#pragma unroll
    for (int rt = 0; rt < ROWS / 16; ++rt) {
      // A operand (16x4 f32): lanes 0-15 hold (K0,K1)=(x,y); lanes 16-31 (z,0)
      const int rr = rt * 16 + m16;
      v2f a;
      a.x = hi ? rowZ[rr] : rowX[rr];
      a.y = hi ? 0.f : rowY[rr];
#pragma unroll
      for (int ct = 0; ct < CCH / 16; ++ct) {
        const int cc = ct * 16 + m16;
        v2f bb;
        bb.x = hi ? colZ[cc] : colX[cc];
        bb.y = hi ? 0.f : colY[cc];
        v8f acc = {};
        acc = __builtin_amdgcn_wmma_f32_16x16x4_f32(
            /*neg_a=*/false, a, /*neg_b=*/false, bb,
            /*c_mod=*/(short)0, acc, /*reuse_a=*/false, /*reuse_b=*/false);
#pragma unroll
        for (int r = 0; r < 8; ++r) {
          const int rl = rt * 16 + r + (hi ? 8 : 0);  // C layout: VGPR r -> M=r / M=r+8
          const int cl = cc;
          const float d2 = rowS[rl] + colS[cl] - 2.f * acc[r];
          float dd = __builtin_amdgcn_sqrtf(fmaxf(d2, 0.f) + SEPSF);
          if ((r0 + rl) == (c0 + cl)) dd += LEPSF;            // eye * LEPS
          const float m2 = rowM[rl] * colM[cl];
          dd = dd * m2 + (1.f - m2) * LEPSF;                  // mask2d
          dtile[rl * DSTRIDE + cl] = dd;
        }
      }
    }
    __syncthreads();

    // prefetch next column chunk while we scan this one (global_prefetch_b8)
    if (c0 + CCH < NATM) {
      const float* nb = &coords[((size_t)b * NATM + (c0 + CCH + tid * 4)) * 3];
      __builtin_prefetch(nb, 0, 0);
    }

    // --- phase B: per-lane top-32 insertion over the tile (b128 scans) -----
    for (int c4 = 0; c4 < CCH; c4 += 4) {
      const float4 dv =
          *reinterpret_cast<const float4*>(&dtile[tid * DSTRIDE + c4]);
      const float mn = fminf(fminf(dv.x, dv.y), fminf(dv.z, dv.w));
      if (mn < kworst) {
        const float q[4] = {dv.x, dv.y, dv.z, dv.w};
#pragma unroll
        for (int e = 0; e < 4; ++e) {
          const float dd = q[e];
          if (dd < kworst) {
            const int j = c0 + c4 + e;
            int p = KNN - 1;
            while (p > 0 && bestD[(p - 1) * ROWS + tid] > dd) {
              bestD[p * ROWS + tid] = bestD[(p - 1) * ROWS + tid];
              bestI[p * ROWS + tid] = bestI[(p - 1) * ROWS + tid];
              --p;
            }
            bestD[p * ROWS + tid] = dd;
            bestI[p * ROWS + tid] = j;
            kworst = bestD[(KNN - 1) * ROWS + tid];
          }
        }
      }
    }
    __syncthreads();
  }

  // --- write results -------------------------------------------------------
  const int gi = r0 + tid;
  const bool pad = (rowM[tid] == 0.f);
  const size_t obase = ((size_t)b * NATM + gi) * KNN;
#pragma unroll
  for (int k = 0; k < KNN; ++k) {
    float dv = bestD[k * ROWS + tid];
    int iv = bestI[k * ROWS + tid];
    if (iv == gi) iv = -1;          // self -> -1
    if (pad) { iv = -1; dv = LEPSF; }
    dists_out[obase + k] = dv;
    edge_out[obase + k]  = iv;
  }
}

// ---------------------------------------------------------------------------
extern "C" void kernel_launch(void* const* d_in, const int* in_sizes, int n_in,
                              void* d_out, int out_size, void* d_ws, size_t ws_size,
                              hipStream_t stream) {
  (void)in_sizes; (void)n_in; (void)out_size; (void)ws_size;
  const float* coords = (const float*)d_in[0];   // [B,N,3]
  const float* mask   = (const float*)d_in[1];   // [B,N]
  const float* table  = (const float*)d_in[2];   // [6,D]
  const float* gamma  = (const float*)d_in[3];   // [D]
  const float* beta   = (const float*)d_in[4];   // [D]

  float* out      = (float*)d_out;
  float* emb_out  = out;                                            // B*N*D
  float* dist_out = out + (size_t)BSZ * NATM * DIM;                 // B*N*K
  int*   edge_out = (int*)(out + (size_t)BSZ * NATM * DIM
                               + (size_t)BSZ * NATM * KNN);         // B*N*K int32
  float* stats = (float*)d_ws;                                      // B*2*D floats

  graphnorm_stats<<<BSZ, 128, 0, stream>>>(mask, table, stats);
  graphnorm_apply<<<dim3(NATM / 8, BSZ), 128, 0, stream>>>(
      mask, table, gamma, beta, stats, emb_out);
  knn_kernel<<<dim3(NATM / ROWS, BSZ), 32, 0, stream>>>(
      coords, mask, dist_out, edge_out);
}